// Decorrelation_35777077576176
// MI455X (gfx1250) — compile-verified
//
#include <hip/hip_runtime.h>

// ---------------------------------------------------------------------------
// Decorrelation as GEMM on the CDNA5 matrix pipe.
//   G[n, k=covar*11+d] = Bernstein(xn[n,covar])_d * x[n,covar]    (built in LDS)
//   W[k, var]          = (covar<var) ? params[d, tril(var,covar)] : 0
//   out = x + G @ W    via V_WMMA_F32_16X16X4_F32 (f32 in / f32 acc)
// ---------------------------------------------------------------------------

typedef __attribute__((ext_vector_type(2))) float v2f;
typedef __attribute__((ext_vector_type(8))) float v8f;

#define DEG        10
#define NC         11            // DEG+1 Bernstein coefficients
#define NV         48            // variables
#define NSAMP      50000
#define NP         1128          // NV*(NV-1)/2
#define KDIM       528           // NV*NC, divisible by 4
#define KSTEPS     132           // KDIM/4
#define NTILES     3             // 48 vars / 16
#define ROWS_BLK   64            // samples per block
#define G_STRIDE   532           // 528 + 4 pad -> conflict-free b64 A-frag reads
#define BLK_THREADS 128          // 4 waves (wave32)

// ---------------------------------------------------------------------------
// Setup kernel: materialize W in the exact per-lane B-fragment order so the
// main loop does one global_load_b64 per (kstep, ntile) per lane.
// B-matrix 4x16 f32 layout (mirror of documented A 16x4 layout):
//   VGPR0: lanes 0-15 -> K=k0,N=lane ; lanes 16-31 -> K=k0+2,N=lane-16
//   VGPR1: same with K+1
// Swizzled element index: ((kk*3 + ntile)*32 + lane) -> float2 {W[k],W[k+1]}
// ---------------------------------------------------------------------------
__global__ void build_w_kernel(const float* __restrict__ params,
                               float* __restrict__ wsw) {
  int tid = blockIdx.x * blockDim.x + threadIdx.x;
  const int total = KSTEPS * NTILES * 32;
  if (tid >= total) return;
  int lane = tid & 31;
  int tile = (tid >> 5) % NTILES;
  int kk   = tid / (32 * NTILES);
  int var  = tile * 16 + (lane & 15);
  int k0   = kk * 4 + ((lane >> 4) ? 2 : 0);
  float vals[2];
#pragma unroll
  for (int i = 0; i < 2; ++i) {
    int k     = k0 + i;
    int covar = k / NC;
    int d     = k - covar * NC;
    float w   = 0.0f;
    if (covar < var)            // strictly lower triangular scatter
      w = params[d * NP + (var * (var - 1)) / 2 + covar];
    vals[i] = w;
  }
  float2* out2 = (float2*)wsw;
  out2[tid] = make_float2(vals[0], vals[1]);
}

// ---------------------------------------------------------------------------
// Main kernel: one block = 64 samples, 4 waves; wave w owns rows [16w,16w+16).
// ---------------------------------------------------------------------------
__global__ void decorrelation_kernel(const float* __restrict__ x,
                                     const float* __restrict__ prange,
                                     const float* __restrict__ wsw,
                                     float* __restrict__ out) {
  extern __shared__ float lds[];              // G: [ROWS_BLK][G_STRIDE] f32
  const int tid     = threadIdx.x;
  const int lane    = tid & 31;
  const int wave    = tid >> 5;
  const int rowBase = blockIdx.x * ROWS_BLK;

  const float binom[NC] = {1.f, 10.f, 45.f, 120.f, 210.f, 252.f,
                           210.f, 120.f, 45.f, 10.f, 1.f};

  // ---- Phase A: build G tile in LDS -------------------------------------
  // 64 rows * 48 covars tasks; consecutive tids -> consecutive covars
  // (coalesced x loads; LDS store addrs differ by 11 floats -> conflict-free).
  for (int task = tid; task < ROWS_BLK * NV; task += BLK_THREADS) {
    int row   = task / NV;
    int covar = task - row * NV;
    int rg    = rowBase + row;
    if (rg >= NSAMP) rg = NSAMP - 1;          // clamp; stores guarded later
    float xv = x[rg * NV + covar];
    float lo = prange[covar];
    float hi = prange[NV + covar];
    float xn = (xv - lo) / (hi - lo);
    float om = 1.0f - xn;
    float p[NC], q[NC];
    p[0] = 1.0f;
    q[0] = 1.0f;
#pragma unroll
    for (int d = 1; d < NC; ++d) {
      p[d] = p[d - 1] * xn;
      q[d] = q[d - 1] * om;
    }
    float* g = &lds[row * G_STRIDE + covar * NC];
#pragma unroll
    for (int d = 0; d < NC; ++d)
      g[d] = binom[d] * p[d] * q[DEG - d] * xv;
  }
  __syncthreads();

  // ---- Phase B: 132 k-steps of V_WMMA_F32_16X16X4_F32 --------------------
  v8f c0 = {}, c1 = {}, c2 = {};
  // A-fragment (16x4 f32): lanes 0-15 rows M=lane K={0,1}; lanes 16-31 K={2,3}
  const int   mrow  = lane & 15;
  const int   khalf = (lane >> 4) * 2;
  const float* aBase = &lds[(wave * 16 + mrow) * G_STRIDE + khalf];
  const v2f*   bBase = (const v2f*)wsw;

#pragma unroll 2
  for (int kk = 0; kk < KSTEPS; ++kk) {
    v2f a  = *(const v2f*)(aBase + kk * 4);           // ds_load_b64, no conflicts
    v2f b0 = bBase[(kk * 3 + 0) * 32 + lane];         // global_load_b64 (L2-hot)
    v2f b1 = bBase[(kk * 3 + 1) * 32 + lane];
    v2f b2 = bBase[(kk * 3 + 2) * 32 + lane];
    c0 = __builtin_amdgcn_wmma_f32_16x16x4_f32(false, a, false, b0,
                                               (short)0, c0, false, false);
    c1 = __builtin_amdgcn_wmma_f32_16x16x4_f32(false, a, false, b1,
                                               (short)0, c1, false, false);
    c2 = __builtin_amdgcn_wmma_f32_16x16x4_f32(false, a, false, b2,
                                               (short)0, c2, false, false);
  }

  // ---- Epilogue: out = x + acc -------------------------------------------
  // C/D f32 16x16 layout: VGPR j -> M=j (lanes 0-15) / M=j+8 (lanes 16-31),
  // N = lane & 15.
  const int col = lane & 15;
  const int rg0 = rowBase + wave * 16 + ((lane >> 4) ? 8 : 0);
#pragma unroll
  for (int j = 0; j < 8; ++j) {
    int rg = rg0 + j;
    if (rg < NSAMP) {
      out[rg * NV + col]      = x[rg * NV + col]      + c0[j];
      out[rg * NV + 16 + col] = x[rg * NV + 16 + col] + c1[j];
      out[rg * NV + 32 + col] = x[rg * NV + 32 + col] + c2[j];
    }
  }
}

// ---------------------------------------------------------------------------
extern "C" void kernel_launch(void* const* d_in, const int* in_sizes, int n_in,
                              void* d_out, int out_size, void* d_ws, size_t ws_size,
                              hipStream_t stream) {
  const float* x      = (const float*)d_in[0];  // [N, V]
  const float* params = (const float*)d_in[1];  // [11, 1128]
  const float* prange = (const float*)d_in[2];  // [2, V]
  float*       out    = (float*)d_out;          // [N, V]
  float*       wsw    = (float*)d_ws;           // swizzled W: 132*3*32 float2 = 101376 B

  // 1) Build swizzled W (12672 float2 entries).
  {
    const int total  = KSTEPS * NTILES * 32;
    const int blocks = (total + 255) / 256;
    build_w_kernel<<<blocks, 256, 0, stream>>>(params, wsw);
  }

  // 2) GEMM + epilogue. 136192 B dynamic LDS per block (<=320KB WGP limit,
  //    2 blocks resident per WGP).
  {
    const int    blocks   = (NSAMP + ROWS_BLK - 1) / ROWS_BLK;   // 782
    const size_t ldsBytes = (size_t)ROWS_BLK * G_STRIDE * sizeof(float);
    decorrelation_kernel<<<blocks, BLK_THREADS, ldsBytes, stream>>>(x, prange,
                                                                    wsw, out);
  }
}